// Mamba2_22497038697218
// MI455X (gfx1250) — compile-verified
//
#include <hip/hip_runtime.h>
#include <hip/hip_bf16.h>
#include <stdint.h>

// ---------------------------------------------------------------------------
// Mamba2 forward for MI455X (gfx1250, wave32).
//   * GEMMs: V_WMMA_F32_16X16X4_F32, LDS-staged tiles, double-buffered with
//     GLOBAL_LOAD_ASYNC_TO_LDS_B128 (ASYNCcnt) when the toolchain exposes it.
//   * Scan: one workgroup per (batch, head), state in registers, per-step
//     B/C/x staged via async-to-LDS double buffering (1 barrier/step).
// ---------------------------------------------------------------------------

typedef __attribute__((ext_vector_type(2))) float v2f;
typedef __attribute__((ext_vector_type(4))) float v4f;
typedef __attribute__((ext_vector_type(8))) float v8f;

namespace cfg {
constexpr int D_MODEL   = 768;
constexpr int D_SSM     = 1536;
constexpr int D_STATE   = 128;
constexpr int HEADDIM   = 64;
constexpr int NHEADS    = D_SSM / HEADDIM;                  // 24
constexpr int CONV_DIM  = D_SSM + 2 * D_STATE;              // 1792
constexpr int D_IN_PROJ = 2 * D_SSM + 2 * D_STATE + NHEADS; // 3352
constexpr int BATCH     = 4;
constexpr int SEQLEN    = 2048;
constexpr int ROWS      = BATCH * SEQLEN;                   // 8192
constexpr float EPS     = 1e-5f;
} // namespace cfg

// -------------------- CDNA5 async global->LDS copy helpers -----------------
#if defined(__gfx1250__) &&                                                    \
    __has_builtin(__builtin_amdgcn_global_load_async_to_lds_b128) &&           \
    __has_builtin(__builtin_amdgcn_s_wait_asynccnt)
#define CDNA5_HAS_ASYNC 1
#else
#define CDNA5_HAS_ASYNC 0
#endif

// Builtin parameter types (from clang diagnostic): global/local int4* .
typedef int vi4 __attribute__((__vector_size__(16)));
typedef __attribute__((address_space(1))) vi4* gvi4_p;   // global (AS1)
typedef __attribute__((address_space(3))) vi4* lvi4_p;   // LDS    (AS3)

// Copy 16 bytes global->LDS. Async when available; sync reg-staged otherwise.
// Generic->AS casts via integer round-trip: global generic == AS1 value;
// low 32 bits of a generic LDS pointer are the LDS byte offset (ISA 10.2).
__device__ __forceinline__ void g2l_b128(void* lds, const void* g)
{
#if CDNA5_HAS_ASYNC
    __builtin_amdgcn_global_load_async_to_lds_b128(
        (gvi4_p)(uintptr_t)g,
        (lvi4_p)(uint32_t)(uintptr_t)lds,
        0, 0);
#else
    *(v4f*)lds = *(const v4f*)g;
#endif
}

__device__ __forceinline__ void async_wait0()
{
#if CDNA5_HAS_ASYNC
    __builtin_amdgcn_s_wait_asynccnt(0);
#endif
}

// ---------------------------------------------------------------------------
// fp32 WMMA GEMM:  C[M,N] = A[M,K] * W[N,K]^T   (all row-major)
// Block: 256 threads = 8 waves (2 M x 4 N); block tile 64(M) x 256(N).
// Wave tile 32x64 (2x4 accumulators of 16x16). K chunked by 16, tiles staged
// in double-buffered LDS (row stride 20 floats -> conflict-free b64 reads).
// A-fragment (16x4): lanes 0-15 M=lane, K={0,1}; lanes 16-31 K={2,3}.
// B-fragment (4x16): mirrored, N=lane&15, K-pair selected by lane>>4.
// C/D 16x16 f32: vgpr g -> M = g + 8*(lane>>4), N = lane&15.
// ---------------------------------------------------------------------------
namespace gcfg {
constexpr int KC   = 16;   // K chunk
constexpr int LSTR = 20;   // padded LDS row stride in floats
} // namespace gcfg

__global__ __launch_bounds__(256)
void wmma_gemm_f32_kernel(const float* __restrict__ A,
                          const float* __restrict__ W,
                          float* __restrict__ C,
                          int M, int N, int K)
{
    using namespace gcfg;
    __shared__ float As[2][64 * LSTR];
    __shared__ float Bs[2][256 * LSTR];

    const int tid  = threadIdx.x;
    const int lane = tid & 31;
    const int wave = tid >> 5;
    const int wm   = wave & 1;
    const int wn   = wave >> 1;
    const int mBlk = blockIdx.x * 64;
    const int nBlk = blockIdx.y * 256;
    const int lrow = lane & 15;
    const int kp   = (lane >> 4) * 2;

    // Global sources for the cooperative tile loads.
    const int arow  = tid >> 2;                     // 0..63
    const int akoff = (tid & 3) * 4;                // 0,4,8,12
    const float* aSrc = A + (size_t)(mBlk + arow) * K + akoff;
    int bn = nBlk + tid;                            // 0..255 -> N rows
    if (bn >= N) bn = N - 1;                        // clamp; cols never stored
    const float* bSrc = W + (size_t)bn * K;

    auto issue = [&](int k0, int buf) {
        g2l_b128(&As[buf][arow * LSTR + akoff], aSrc + k0);
        #pragma unroll
        for (int q = 0; q < 4; ++q)
            g2l_b128(&Bs[buf][tid * LSTR + q * 4], bSrc + k0 + q * 4);
    };

    v8f acc[2][4];
    #pragma unroll
    for (int i = 0; i < 2; ++i)
        #pragma unroll
        for (int j = 0; j < 4; ++j)
            acc[i][j] = (v8f)0.0f;

    const int nk = K / KC;
    issue(0, 0);
    async_wait0();
    __syncthreads();

    for (int kc = 0; kc < nk; ++kc) {
        const int buf = kc & 1;
        if (kc + 1 < nk) issue((kc + 1) * KC, buf ^ 1);

        const float* Ab = &As[buf][0];
        const float* Bb = &Bs[buf][0];
        #pragma unroll
        for (int kk = 0; kk < KC; kk += 4) {
            v2f a[2], b[4];
            #pragma unroll
            for (int i = 0; i < 2; ++i) {
                const int r = (wm * 32 + i * 16 + lrow) * LSTR + kk + kp;
                a[i].x = Ab[r];
                a[i].y = Ab[r + 1];
            }
            #pragma unroll
            for (int j = 0; j < 4; ++j) {
                const int r = (wn * 64 + j * 16 + lrow) * LSTR + kk + kp;
                b[j].x = Bb[r];
                b[j].y = Bb[r + 1];
            }
            #pragma unroll
            for (int i = 0; i < 2; ++i)
                #pragma unroll
                for (int j = 0; j < 4; ++j)
                    acc[i][j] = __builtin_amdgcn_wmma_f32_16x16x4_f32(
                        false, a[i], false, b[j], (short)0, acc[i][j],
                        false, false);
        }
        async_wait0();   // chunk kc+1 resident (overlapped with compute above)
        __syncthreads(); // + safe to overwrite buf kc in next iteration
    }

    const int srow = (lane >> 4) * 8;               // 0 or 8
    #pragma unroll
    for (int j = 0; j < 4; ++j) {
        const int n = nBlk + wn * 64 + j * 16 + lrow;
        if (n >= N) continue;
        #pragma unroll
        for (int i = 0; i < 2; ++i) {
            #pragma unroll
            for (int g = 0; g < 8; ++g) {
                const int m = mBlk + wm * 32 + i * 16 + srow + g;
                C[(size_t)m * N + n] = acc[i][j][g];
            }
        }
    }
}

// ---------------------------------------------------------------------------
// Depthwise causal conv (width 4) + bias + SiLU over the xBC slice of zxbcdt.
// zxbcdt row layout: [0,1536)=z  [1536,3328)=xBC  [3328,3352)=dt_raw
// ---------------------------------------------------------------------------
__global__ __launch_bounds__(256)
void conv_silu_kernel(const float* __restrict__ zx,
                      const float* __restrict__ cw,   // [CONV_DIM, 4]
                      const float* __restrict__ cb,   // [CONV_DIM]
                      float* __restrict__ xBC)        // [B, L, CONV_DIM]
{
    using namespace cfg;
    const long long idx = (long long)blockIdx.x * blockDim.x + threadIdx.x;
    const long long total = (long long)BATCH * SEQLEN * CONV_DIM;
    if (idx >= total) return;
    const int c = (int)(idx % CONV_DIM);
    const int l = (int)((idx / CONV_DIM) % SEQLEN);
    const int b = (int)(idx / ((long long)CONV_DIM * SEQLEN));

    const float w0 = cw[c * 4 + 0], w1 = cw[c * 4 + 1];
    const float w2 = cw[c * 4 + 2], w3 = cw[c * 4 + 3];
    const size_t base = (size_t)b * SEQLEN * D_IN_PROJ + (size_t)D_SSM + c;

    float acc = cb[c];
    if (l >= 3) acc += zx[base + (size_t)(l - 3) * D_IN_PROJ] * w0;
    if (l >= 2) acc += zx[base + (size_t)(l - 2) * D_IN_PROJ] * w1;
    if (l >= 1) acc += zx[base + (size_t)(l - 1) * D_IN_PROJ] * w2;
    acc += zx[base + (size_t)l * D_IN_PROJ] * w3;

    xBC[idx] = acc / (1.0f + __expf(-acc));          // SiLU
}

// ---------------------------------------------------------------------------
// dt = softplus(dt_raw + dt_bias)  -> [B*L, NHEADS]
// ---------------------------------------------------------------------------
__global__ __launch_bounds__(256)
void dt_softplus_kernel(const float* __restrict__ zx,
                        const float* __restrict__ dt_bias,
                        float* __restrict__ dt_out)
{
    using namespace cfg;
    const int idx = blockIdx.x * blockDim.x + threadIdx.x;
    if (idx >= ROWS * NHEADS) return;
    const int h = idx % NHEADS;
    const long long row = idx / NHEADS;
    const float v = zx[(size_t)row * D_IN_PROJ + (D_SSM + CONV_DIM) + h] + dt_bias[h];
    dt_out[idx] = (v > 20.0f) ? v : log1pf(__expf(v));
}

// ---------------------------------------------------------------------------
// Selective scan. One block (256 thr = 8 wave32) per (b, head).
// Thread owns p = tid>>2 (0..63) and n in [(tid&3)*32, +32): 32 fp32 state
// regs. Per-step B/C/x (320 floats) double-buffered in LDS via async copy:
// step l+1 streams in while step l computes; single barrier per step.
// sbuf layout: [0,128)=B  [128,256)=C  [256,320)=x
// ---------------------------------------------------------------------------
__global__ __launch_bounds__(256)
void scan_kernel(const float* __restrict__ xBC,   // [B,L,CONV_DIM]
                 const float* __restrict__ dt,    // [B,L,NHEADS]
                 const float* __restrict__ A_log, // [NHEADS]
                 const float* __restrict__ Dp,    // [NHEADS]
                 float* __restrict__ y)           // [B,L,D_SSM]
{
    using namespace cfg;
    const int b   = blockIdx.x / NHEADS;
    const int hd  = blockIdx.x % NHEADS;
    const int tid = threadIdx.x;
    const int p   = tid >> 2;
    const int nb  = (tid & 3) * 32;

    const float Ah = -__expf(A_log[hd]);
    const float Dh = Dp[hd];

    __shared__ float sbuf[2][2 * D_STATE + HEADDIM];   // [2][320]

    float h[32];
    #pragma unroll
    for (int j = 0; j < 32; ++j) h[j] = 0.0f;

    const size_t bBase = (size_t)b * SEQLEN;

    auto issue = [&](int l, int buf) {
        if (tid < 80) {
            const float* xrow = xBC + (bBase + l) * CONV_DIM;
            const float* src = (tid < 64)
                                   ? xrow + D_SSM + tid * 4                  // B,C
                                   : xrow + hd * HEADDIM + (tid - 64) * 4;   // x
            g2l_b128(&sbuf[buf][tid * 4], src);
        }
    };

    issue(0, 0);
    async_wait0();
    __syncthreads();

    for (int l = 0; l < SEQLEN; ++l) {
        const int buf = l & 1;
        if (l + 1 < SEQLEN) issue(l + 1, buf ^ 1);

        const float* Bsh = &sbuf[buf][0];
        const float* Csh = Bsh + D_STATE;
        const float* xsh = Bsh + 2 * D_STATE;

        const float dtv = dt[(bBase + l) * NHEADS + hd];
        const float dA  = __expf(dtv * Ah);
        const float s   = dtv * xsh[p];

        float yacc = 0.0f;
        #pragma unroll
        for (int j = 0; j < 32; ++j) {
            h[j] = h[j] * dA + s * Bsh[nb + j];
            yacc += h[j] * Csh[nb + j];
        }
        yacc += __shfl_xor(yacc, 1, 32);
        yacc += __shfl_xor(yacc, 2, 32);

        if ((tid & 3) == 0)
            y[(bBase + l) * D_SSM + hd * HEADDIM + p] = yacc + Dh * xsh[p];

        async_wait0();   // step l+1 staged (overlapped with compute above)
        __syncthreads(); // reads of buf l done; next iter overwrites it
    }
}

// ---------------------------------------------------------------------------
// yn = (y * silu(z)) * rsqrt(mean((y*silu(z))^2) + eps) * norm_w
// One block per row (b*L+l); 1536 elems -> 6 per thread.
// ---------------------------------------------------------------------------
__global__ __launch_bounds__(256)
void gate_rmsnorm_kernel(const float* __restrict__ y,
                         const float* __restrict__ zx,
                         const float* __restrict__ norm_w,
                         float* __restrict__ yn)
{
    using namespace cfg;
    const int row = blockIdx.x;
    const int tid = threadIdx.x;
    const float* zrow = zx + (size_t)row * D_IN_PROJ;
    const float* yrow = y  + (size_t)row * D_SSM;

    float g[6];
    float ss = 0.0f;
    #pragma unroll
    for (int it = 0; it < 6; ++it) {
        const int i = tid + it * 256;
        const float zv = zrow[i];
        const float yv = yrow[i];
        const float gv = yv * (zv / (1.0f + __expf(-zv)));
        g[it] = gv;
        ss += gv * gv;
    }
    #pragma unroll
    for (int off = 16; off > 0; off >>= 1) ss += __shfl_xor(ss, off, 32);

    __shared__ float red[8];
    if ((tid & 31) == 0) red[tid >> 5] = ss;
    __syncthreads();
    float tot = 0.0f;
    #pragma unroll
    for (int w = 0; w < 8; ++w) tot += red[w];
    const float scale = rsqrtf(tot / (float)D_SSM + EPS);

    float* out = yn + (size_t)row * D_SSM;
    #pragma unroll
    for (int it = 0; it < 6; ++it) {
        const int i = tid + it * 256;
        out[i] = g[it] * scale * norm_w[i];
    }
}

// ---------------------------------------------------------------------------
// Launch
// ---------------------------------------------------------------------------
extern "C" void kernel_launch(void* const* d_in, const int* in_sizes, int n_in,
                              void* d_out, int out_size, void* d_ws, size_t ws_size,
                              hipStream_t stream)
{
    using namespace cfg;
    const float* u       = (const float*)d_in[0];
    const float* W_in    = (const float*)d_in[1];
    const float* conv_w  = (const float*)d_in[2];
    const float* conv_b  = (const float*)d_in[3];
    const float* dt_bias = (const float*)d_in[4];
    const float* A_log   = (const float*)d_in[5];
    const float* Dp      = (const float*)d_in[6];
    const float* norm_w  = (const float*)d_in[7];
    const float* W_out   = (const float*)d_in[8];
    float* out = (float*)d_out;

    // Workspace layout (floats)
    float* ws     = (float*)d_ws;
    float* zx     = ws;                                  // [ROWS, 3352]
    float* xBCact = zx + (size_t)ROWS * D_IN_PROJ;       // [ROWS, 1792]
    float* dtbuf  = xBCact + (size_t)ROWS * CONV_DIM;    // [ROWS, 24]
    float* ybuf   = dtbuf + (size_t)ROWS * NHEADS;       // [ROWS, 1536]
    float* ynbuf  = ybuf + (size_t)ROWS * D_SSM;         // [ROWS, 1536]

    // 1) in_proj GEMM: zx = u @ W_in^T   (M=8192, N=3352, K=768)
    {
        dim3 grid(ROWS / 64, (D_IN_PROJ + 255) / 256);
        wmma_gemm_f32_kernel<<<grid, 256, 0, stream>>>(u, W_in, zx,
                                                       ROWS, D_IN_PROJ, D_MODEL);
    }
    // 2) depthwise conv + SiLU
    {
        const long long total = (long long)ROWS * CONV_DIM;
        conv_silu_kernel<<<(int)((total + 255) / 256), 256, 0, stream>>>(
            zx, conv_w, conv_b, xBCact);
    }
    // 3) dt = softplus(dt_raw + bias)
    {
        const int total = ROWS * NHEADS;
        dt_softplus_kernel<<<(total + 255) / 256, 256, 0, stream>>>(
            zx, dt_bias, dtbuf);
    }
    // 4) selective scan (+ D*x fused)
    scan_kernel<<<BATCH * NHEADS, 256, 0, stream>>>(xBCact, dtbuf, A_log, Dp, ybuf);

    // 5) gate + RMSNorm
    gate_rmsnorm_kernel<<<ROWS, 256, 0, stream>>>(ybuf, zx, norm_w, ynbuf);

    // 6) out_proj GEMM: out = yn @ W_out^T  (M=8192, N=768, K=1536)
    {
        dim3 grid(ROWS / 64, (D_MODEL + 255) / 256);
        wmma_gemm_f32_kernel<<<grid, 256, 0, stream>>>(ynbuf, W_out, out,
                                                       ROWS, D_MODEL, D_SSM);
    }
    (void)in_sizes; (void)n_in; (void)out_size; (void)ws_size;
}